// my_param_layer_60842506715778
// MI455X (gfx1250) — compile-verified
//
#include <hip/hip_runtime.h>

typedef __attribute__((ext_vector_type(2))) float v2f;
typedef __attribute__((ext_vector_type(4))) float v4f;
typedef __attribute__((ext_vector_type(8))) float v8f;
typedef __attribute__((ext_vector_type(4))) unsigned int u32x4;
typedef __attribute__((ext_vector_type(8))) int i32x8;
typedef __attribute__((ext_vector_type(4))) int i32x4;

#define AS1 __attribute__((address_space(1)))
#define AS3 __attribute__((address_space(3)))

#define MDIM 8000
#define KDIM 120
#define NDIM 8000
#define MPAD 8064
#define KP   128     // padded K
#define TM   128     // C-tile rows per block
#define TN   64      // C-tile cols per block

#if defined(__gfx1250__) && __has_builtin(__builtin_amdgcn_tensor_load_to_lds)
#define USE_TDM 1
#elif defined(__gfx1250__) && __has_builtin(__builtin_amdgcn_global_load_async_to_lds_b128)
#define USE_ASYNC 1
#endif

#ifdef USE_TDM
// ---- Tensor DMA descriptor packing (CDNA5 ISA) ----
// Group 0: [1:0]=count(1), [63:32]=lds_addr, [120:64]=global_addr, [127:126]=type(2)
__device__ __forceinline__ u32x4 tdm_g0(unsigned lds_byte, unsigned long long gaddr) {
    u32x4 g;
    g.x = 1u;                                        // count=1, user mode, no gather
    g.y = lds_byte;                                  // LDS byte address
    g.z = (unsigned)(gaddr & 0xFFFFFFFFu);           // global_addr[31:0]
    g.w = (unsigned)((gaddr >> 32) & 0x01FFFFFFu)    // global_addr[56:32]
        | (2u << 30);                                // type = 2 ("image")
    return g;
}
// Group 1: [17:16]=data_size(2 -> 4B), [79:48]=tensor_dim0, [111:80]=tensor_dim1,
// [127:112]=tile_dim0, [143:128]=tile_dim1, [159:144]=tile_dim2(0),
// [207:160]=tensor_dim0_stride, [255:208]=tensor_dim1_stride(0, 2D tile)
__device__ __forceinline__ i32x8 tdm_g1(unsigned t_d0, unsigned t_d1,
                                        unsigned tile_d0, unsigned tile_d1,
                                        unsigned long long stride0) {
    i32x8 g = {0, 0, 0, 0, 0, 0, 0, 0};
    g[0] = (int)(2u << 16);                                          // data_size=4B
    g[1] = (int)((t_d0 & 0xFFFFu) << 16);                            // dim0[15:0]
    g[2] = (int)(((t_d0 >> 16) & 0xFFFFu) | ((t_d1 & 0xFFFFu) << 16));
    g[3] = (int)(((t_d1 >> 16) & 0xFFFFu) | ((tile_d0 & 0xFFFFu) << 16));
    g[4] = (int)(tile_d1 & 0xFFFFu);                                 // tile_dim2 = 0
    g[5] = (int)(stride0 & 0xFFFFFFFFu);
    g[6] = (int)((stride0 >> 32) & 0xFFFFu);                         // dim1_stride = 0
    g[7] = 0;
    return g;
}
#endif

// 256 threads = 8 waves (wave32). Wave grid 4x2: each wave owns a 32x32 C
// region = 2x2 WMMA 16x16 accumulators. One LDS fill covers all of K (=128).
__global__ __launch_bounds__(256) void gemm_bias_wmma_f32(
    const float* __restrict__ A,    // [8000,120] row-major
    const float* __restrict__ B,    // [120,8000] row-major
    const float* __restrict__ bias, // [8000]
    float* __restrict__ out)        // [8064,8000] row-major
{
    __shared__ float lsA[TM * KP];  // 64 KB, row-major [m][k]
    __shared__ float lsB[KP * TN];  // 32 KB, row-major [k][n]

    const int tid  = threadIdx.x;
    const int lane = tid & 31;
    const int wave = tid >> 5;
    const int n0   = blockIdx.x * TN;   // 125 tiles
    const int m0   = blockIdx.y * TM;   // 63 tiles

#if defined(USE_TDM)
    // One TDM descriptor per tile; OOB rows/cols (k>=120, m>=8000) read as zero,
    // which implements the reference's zero-padding for free. Wave 0 issues both.
    if (tid < 32) {
        const unsigned ldsA = (unsigned)(unsigned long long)(AS3 float*)lsA;
        const unsigned ldsB = (unsigned)(unsigned long long)(AS3 float*)lsB;
        // A tile: x=k (tile 128, valid 120), y=m (tile 128, valid MDIM-m0), stride 120
        u32x4 a0 = tdm_g0(ldsA, (unsigned long long)(A + (long)m0 * KDIM));
        i32x8 a1 = tdm_g1(KDIM, (unsigned)(MDIM - m0), KP, TM, KDIM);
        // B tile: x=n (tile 64), y=k (tile 128, valid 120), stride 8000
        u32x4 b0 = tdm_g0(ldsB, (unsigned long long)(B + n0));
        i32x8 b1 = tdm_g1(TN, KDIM, TN, KP, NDIM);
        i32x4 z4 = {0, 0, 0, 0};
#if __clang_major__ >= 23
        i32x8 z8 = {0, 0, 0, 0, 0, 0, 0, 0};
        __builtin_amdgcn_tensor_load_to_lds(a0, a1, z4, z4, z8, 0);
        __builtin_amdgcn_tensor_load_to_lds(b0, b1, z4, z4, z8, 0);
#else
        __builtin_amdgcn_tensor_load_to_lds(a0, a1, z4, z4, 0);
        __builtin_amdgcn_tensor_load_to_lds(b0, b1, z4, z4, 0);
#endif
        __builtin_amdgcn_s_wait_tensorcnt(0);
    }
    __syncthreads();
#elif defined(USE_ASYNC)
    // Async global->LDS (ASYNCcnt path); zero-fill padded regions with DS stores.
    {
        const int kv = tid & 31, r0 = tid >> 5, k = kv * 4;
        #pragma unroll
        for (int p = 0; p < 16; ++p) {
            const int r = r0 + p * 8, gr = m0 + r;
            if (gr < MDIM && k < KDIM)
                __builtin_amdgcn_global_load_async_to_lds_b128(
                    (AS1 void*)(A + (long)gr * KDIM + k),
                    (AS3 void*)(lsA + r * KP + k), 0, 0);
            else
                *(v4f*)(lsA + r * KP + k) = (v4f){0.f, 0.f, 0.f, 0.f};
        }
    }
    {
        const int nv = tid & 15, kr0 = tid >> 4, n = nv * 4;
        #pragma unroll
        for (int p = 0; p < 8; ++p) {
            const int kr = kr0 + p * 16;
            if (kr < KDIM)
                __builtin_amdgcn_global_load_async_to_lds_b128(
                    (AS1 void*)(B + (long)kr * NDIM + n0 + n),
                    (AS3 void*)(lsB + kr * TN + n), 0, 0);
            else
                *(v4f*)(lsB + kr * TN + n) = (v4f){0.f, 0.f, 0.f, 0.f};
        }
    }
#if __has_builtin(__builtin_amdgcn_s_wait_asynccnt)
    __builtin_amdgcn_s_wait_asynccnt(0);
#else
    asm volatile("s_wait_asynccnt 0" ::: "memory");
#endif
    __syncthreads();
#else
    // ---- synchronous fallback fill (known-good) ----
    {
        const int kv = tid & 31, r0 = tid >> 5, k = kv * 4;
        #pragma unroll
        for (int p = 0; p < 16; ++p) {
            const int r = r0 + p * 8, gr = m0 + r;
            v4f val = {0.f, 0.f, 0.f, 0.f};
            if (gr < MDIM && k < KDIM)
                val = *(const v4f*)(A + (long)gr * KDIM + k);
            *(v4f*)(lsA + r * KP + k) = val;
        }
    }
    {
        const int nv = tid & 15, kr0 = tid >> 4, n = nv * 4;
        #pragma unroll
        for (int p = 0; p < 8; ++p) {
            const int kr = kr0 + p * 16;
            v4f val = {0.f, 0.f, 0.f, 0.f};
            if (kr < KDIM)
                val = *(const v4f*)(B + (long)kr * NDIM + n0 + n);
            *(v4f*)(lsB + kr * TN + n) = val;
        }
    }
    __syncthreads();
#endif

    const int waveM = (wave & 3) * 32;  // 0,32,64,96
    const int waveN = (wave >> 2) * 32; // 0,32
    const int half  = lane >> 4;        // lane-half selects K pair
    const int l16   = lane & 15;

    v8f acc[2][2] = {};
    const int mrow[2] = { waveM + l16, waveM + 16 + l16 };
    const int ncol[2] = { waveN + l16, waveN + 16 + l16 };

    // K loop: 32 x V_WMMA_F32_16X16X4_F32 per accumulator tile
    for (int k0 = 0; k0 < KP; k0 += 4) {
        const int ka = k0 + 2 * half;   // this lane-half's K pair
        v2f a0 = *(const v2f*)(lsA + mrow[0] * KP + ka);
        v2f a1 = *(const v2f*)(lsA + mrow[1] * KP + ka);
        v2f b0, b1;
        b0.x = lsB[ ka      * TN + ncol[0]];
        b0.y = lsB[(ka + 1) * TN + ncol[0]];
        b1.x = lsB[ ka      * TN + ncol[1]];
        b1.y = lsB[(ka + 1) * TN + ncol[1]];

        acc[0][0] = __builtin_amdgcn_wmma_f32_16x16x4_f32(
            false, a0, false, b0, (short)0, acc[0][0], false, false);
        acc[0][1] = __builtin_amdgcn_wmma_f32_16x16x4_f32(
            false, a0, false, b1, (short)0, acc[0][1], false, false);
        acc[1][0] = __builtin_amdgcn_wmma_f32_16x16x4_f32(
            false, a1, false, b0, (short)0, acc[1][0], false, false);
        acc[1][1] = __builtin_amdgcn_wmma_f32_16x16x4_f32(
            false, a1, false, b1, (short)0, acc[1][1], false, false);
    }

    // ---- store C + bias: C layout row = 8*half + v, col = l16 ----
    #pragma unroll
    for (int nt = 0; nt < 2; ++nt) {
        const int gc = n0 + ncol[nt];
        const float bv = bias[gc];
        #pragma unroll
        for (int mt = 0; mt < 2; ++mt) {
            const int grb = m0 + waveM + mt * 16 + 8 * half;
            #pragma unroll
            for (int v = 0; v < 8; ++v) {
                out[(long)(grb + v) * NDIM + gc] = acc[mt][nt][v] + bv;
            }
        }
    }
}

extern "C" void kernel_launch(void* const* d_in, const int* in_sizes, int n_in,
                              void* d_out, int out_size, void* d_ws, size_t ws_size,
                              hipStream_t stream) {
    const float* A    = (const float*)d_in[0];
    const float* B    = (const float*)d_in[1];
    const float* bias = (const float*)d_in[2];
    float* out        = (float*)d_out;

    dim3 grid(NDIM / TN, MPAD / TM);   // (125, 63)
    dim3 block(256);
    gemm_bias_wmma_f32<<<grid, block, 0, stream>>>(A, B, bias, out);
}